// GCN2_25159918420550
// MI455X (gfx1250) — compile-verified
//
#include <hip/hip_runtime.h>

typedef __attribute__((ext_vector_type(2))) float v2f;
typedef __attribute__((ext_vector_type(8))) float v8f;

#define EPSV 1e-5f

// ---------------------------------------------------------------- utilities
__global__ void k_fill_zero4(float4* __restrict__ p, int n4) {
  int i = blockIdx.x * blockDim.x + threadIdx.x;
  if (i < n4) p[i] = make_float4(0.0f, 0.0f, 0.0f, 0.0f);
}

__global__ void k_degree(const int* __restrict__ src, const int* __restrict__ dst,
                         float* __restrict__ degs, float* __restrict__ degd, int E) {
  int e = blockIdx.x * blockDim.x + threadIdx.x;
  if (e < E) {
    atomicAdd(&degs[src[e]], 1.0f);
    atomicAdd(&degd[dst[e]], 1.0f);
  }
}

__global__ void k_isqrt(float* __restrict__ p, int n) {
  int i = blockIdx.x * blockDim.x + threadIdx.x;
  if (i < n) p[i] = rsqrtf(fmaxf(p[i], 1.0f));
}

// ---------------------------------------------------------------- fp32 WMMA GEMM
// T[nrows x F] = H[nrows x K] @ W[K x F], row-major, exact fp32 via
// V_WMMA_F32_16X16X4_F32. One wave32 owns a 16-row stripe and FT column tiles.
//
// A 16x4 f32 layout : lane l -> row l%16, K = 2*(l/16)+{0,1} in v2f {x,y}
// B 4x16 f32 layout : lane l -> col l%16, K = 2*(l/16)+{0,1} in v2f {x,y}
// C/D 16x16 layout  : VGPR v, lane l -> row v + 8*(l/16), col l%16
template<int K, int F, int FT>
__global__ void k_gemm_wmma(const float* __restrict__ H, const float* __restrict__ W,
                            float* __restrict__ T, int nrows) {
  const int lane = threadIdx.x & 31;
  const int wave = blockIdx.x * (blockDim.x >> 5) + (threadIdx.x >> 5);
  const int row0 = wave << 4;
  if (row0 >= nrows) return;             // uniform per wave: EXEC stays all-1s
  const int rl = lane & 15;
  const int kh = (lane >> 4) << 1;       // 0 or 2

  int row = row0 + rl;
  if (row >= nrows) row = nrows - 1;     // clamp read; stores are guarded
  v8f acc[FT] = {};
  const float* hp = H + (size_t)row * K + kh;

  for (int k0 = 0; k0 < K; k0 += 4) {
    v2f a;
    a.x = hp[k0];
    a.y = hp[k0 + 1];
#pragma unroll
    for (int t = 0; t < FT; ++t) {
      const int col = t * 16 + rl;
      v2f b;
      if (col < F) {
        b.x = W[(size_t)(k0 + kh) * F + col];
        b.y = W[(size_t)(k0 + kh + 1) * F + col];
      } else {
        b.x = 0.0f; b.y = 0.0f;
      }
      acc[t] = __builtin_amdgcn_wmma_f32_16x16x4_f32(
          /*neg_a=*/false, a, /*neg_b=*/false, b,
          /*c_mod=*/(short)0, acc[t], /*reuse_a=*/false, /*reuse_b=*/false);
    }
  }

  const int rbase = row0 + ((lane >> 4) << 3);
  if (row0 + 16 <= nrows) {
    // fast path (always taken when nrows % 16 == 0): no per-row guard,
    // single per-lane column mask covering 8 stores.
#pragma unroll
    for (int t = 0; t < FT; ++t) {
      const int col = t * 16 + rl;
      if (col < F) {
        float* tp = T + (size_t)rbase * F + col;
#pragma unroll
        for (int v = 0; v < 8; ++v) tp[(size_t)v * F] = acc[t][v];
      }
    }
  } else {
#pragma unroll
    for (int t = 0; t < FT; ++t) {
      const int col = t * 16 + rl;
#pragma unroll
      for (int v = 0; v < 8; ++v) {
        const int r = rbase + v;
        if (r < nrows && col < F) T[(size_t)r * F + col] = acc[t][v];
      }
    }
  }
}

// ---------------------------------------------------------------- edge scatter (F = 64, float4 gather)
// acc[dst[e], f] += h[src[e], f] * osq[src[e]] — the bandwidth-bound phase.
// 16 threads per edge, each gathers a float4 (global_load_b128) and issues
// 4 f32 atomic adds (atomics stay scalar for collision correctness).
__global__ void k_scatter64_v4(const float* __restrict__ h, const int* __restrict__ src,
                               const int* __restrict__ dst, const float* __restrict__ osq,
                               float* __restrict__ acc, int E) {
  int i = blockIdx.x * blockDim.x + threadIdx.x;
  int total = E << 4;                    // E * 16 chunks
  if (i >= total) return;
  int e  = i >> 4;
  int f4 = (i & 15) << 2;
  int s = src[e];
  int d = dst[e];
  float sc = osq[s];
  float4 v = *(const float4*)(h + (size_t)s * 64 + f4);
  float* ap = acc + (size_t)d * 64 + f4;
  atomicAdd(ap + 0, v.x * sc);
  atomicAdd(ap + 1, v.y * sc);
  atomicAdd(ap + 2, v.z * sc);
  atomicAdd(ap + 3, v.w * sc);
}

// scalar scatter for F=47 (rows not 16B-aligned)
__global__ void k_scatter(const float* __restrict__ h, const int* __restrict__ src,
                          const int* __restrict__ dst, const float* __restrict__ osq,
                          float* __restrict__ acc, int E, int F) {
  int i = blockIdx.x * blockDim.x + threadIdx.x;
  int total = E * F;
  if (i >= total) return;
  int e = i / F;
  int f = i - e * F;
  int s = src[e];
  float v = h[(size_t)s * F + f] * osq[s];
  atomicAdd(&acc[(size_t)dst[e] * F + f], v);
}

// ---------------------------------------------------------------- fused finalize + BN stats (F = 64)
// h = h * in_isqrt[row] + bias (in place), accumulate column sum / sumsq.
__global__ void k_finalize_stats(float* __restrict__ h, const float* __restrict__ isq,
                                 const float* __restrict__ bias, float* __restrict__ stats,
                                 int nrows) {
  __shared__ float ss[256];
  __shared__ float sq[256];
  const int c = threadIdx.x & 63;
  const int rs = threadIdx.x >> 6;           // 4 row-lanes per block
  const float bc = bias[c];
  float lsum = 0.0f, lsq = 0.0f;
  for (int r = blockIdx.x * 4 + rs; r < nrows; r += gridDim.x * 4) {
    size_t idx = (size_t)r * 64 + c;
    float v = h[idx] * isq[r] + bc;
    h[idx] = v;
    lsum += v;
    lsq += v * v;
  }
  ss[threadIdx.x] = lsum;
  sq[threadIdx.x] = lsq;
  __syncthreads();
  if (rs == 0) {
    float s = ss[c] + ss[c + 64] + ss[c + 128] + ss[c + 192];
    float q = sq[c] + sq[c + 64] + sq[c + 128] + sq[c + 192];
    atomicAdd(&stats[c], s);
    atomicAdd(&stats[64 + c], q);
  }
}

// ---------------------------------------------------------------- BN(affine) + ReLU (F = 64)
__global__ void k_bn_relu(float* __restrict__ h, const float* __restrict__ stats,
                          const float* __restrict__ gamma, const float* __restrict__ beta,
                          int nrows) {
  int i = blockIdx.x * blockDim.x + threadIdx.x;
  int total = nrows * 64;
  if (i >= total) return;
  int c = i & 63;
  float inv_n = 1.0f / (float)nrows;
  float mu = stats[c] * inv_n;
  float var = stats[64 + c] * inv_n - mu * mu;     // biased variance, like torch BN
  float v = gamma[c] * (h[i] - mu) * rsqrtf(var + EPSV) + beta[c];
  h[i] = v > 0.0f ? v : 0.0f;
}

// ---------------------------------------------------------------- output finalize (F = 47)
__global__ void k_finalize_out(float* __restrict__ out, const float* __restrict__ isq,
                               const float* __restrict__ bias, int nrows, int F) {
  int i = blockIdx.x * blockDim.x + threadIdx.x;
  int total = nrows * F;
  if (i >= total) return;
  int r = i / F;
  int c = i - r * F;
  out[i] = out[i] * isq[r] + bias[c];
}

// ---------------------------------------------------------------- launch
extern "C" void kernel_launch(void* const* d_in, const int* in_sizes, int n_in,
                              void* d_out, int out_size, void* d_ws, size_t ws_size,
                              hipStream_t stream) {
  const float* X   = (const float*)d_in[0];
  const int*   src = (const int*)d_in[1];
  const int*   dst = (const int*)d_in[2];
  const float* W0  = (const float*)d_in[3];
  const float* b0  = (const float*)d_in[4];
  const float* W1  = (const float*)d_in[5];
  const float* b1  = (const float*)d_in[6];
  const float* W2  = (const float*)d_in[7];
  const float* b2  = (const float*)d_in[8];
  const float* g0  = (const float*)d_in[9];
  const float* be0 = (const float*)d_in[10];
  const float* g1  = (const float*)d_in[11];
  const float* be1 = (const float*)d_in[12];
  float* out = (float*)d_out;

  const int N = in_sizes[0] / 128;   // nodes
  const int E = in_sizes[1];         // edges

  // workspace layout (floats): bufA[N*64] | bufB[N*64] | out_isqrt[N] | in_isqrt[N] | stats[128]
  float* ws    = (float*)d_ws;
  float* bufA  = ws;
  float* bufB  = bufA + (size_t)N * 64;
  float* iso   = bufB + (size_t)N * 64;
  float* isi   = iso + N;
  float* stats = isi + N;

  const int TPB = 256;
  auto gsz = [](int n, int t) { return (n + t - 1) / t; };

  // degree -> D^{-1/2} (out from src, in from dst); clamp deg>=1 like DGL
  k_fill_zero4<<<gsz(2 * N / 4, TPB), TPB, 0, stream>>>((float4*)iso, 2 * N / 4);
  k_degree<<<gsz(E, TPB), TPB, 0, stream>>>(src, dst, iso, isi, E);
  k_isqrt<<<gsz(2 * N, TPB), TPB, 0, stream>>>(iso, 2 * N);

  const int waves = gsz(N, 16);
  const int gemmBlocks = gsz(waves, 4);   // 128 threads = 4 wave32 per block

  // ---- layer 0 (128 -> 64): matmul first (shrinking W), then aggregate
  k_gemm_wmma<128, 64, 4><<<gemmBlocks, 128, 0, stream>>>(X, W0, bufA, N);
  k_fill_zero4<<<gsz(N * 16, TPB), TPB, 0, stream>>>((float4*)bufB, N * 16);
  k_scatter64_v4<<<gsz(E * 16, TPB), TPB, 0, stream>>>(bufA, src, dst, iso, bufB, E);
  k_fill_zero4<<<1, 32, 0, stream>>>((float4*)stats, 32);
  k_finalize_stats<<<1024, 256, 0, stream>>>(bufB, isi, b0, stats, N);
  k_bn_relu<<<gsz(N * 64, TPB), TPB, 0, stream>>>(bufB, stats, g0, be0, N);

  // ---- layer 1 (64 -> 64): aggregate first, then matmul (DGL order)
  k_fill_zero4<<<gsz(N * 16, TPB), TPB, 0, stream>>>((float4*)bufA, N * 16);
  k_scatter64_v4<<<gsz(E * 16, TPB), TPB, 0, stream>>>(bufB, src, dst, iso, bufA, E);
  k_gemm_wmma<64, 64, 4><<<gemmBlocks, 128, 0, stream>>>(bufA, W1, bufB, N);
  k_fill_zero4<<<1, 32, 0, stream>>>((float4*)stats, 32);
  k_finalize_stats<<<1024, 256, 0, stream>>>(bufB, isi, b1, stats, N);
  k_bn_relu<<<gsz(N * 64, TPB), TPB, 0, stream>>>(bufB, stats, g1, be1, N);

  // ---- layer 2 (64 -> 47): matmul first, aggregate directly into d_out
  k_gemm_wmma<64, 47, 3><<<gemmBlocks, 128, 0, stream>>>(bufB, W2, bufA, N);
  k_fill_zero4<<<gsz(N * 47 / 4, TPB), TPB, 0, stream>>>((float4*)out, N * 47 / 4);
  k_scatter<<<gsz(E * 47, TPB), TPB, 0, stream>>>(bufA, src, dst, iso, out, E, 47);
  k_finalize_out<<<gsz(N * 47, TPB), TPB, 0, stream>>>(out, isi, b2, N, 47);
}